// ForwardDiffusionModel_26895085207689
// MI455X (gfx1250) — compile-verified
//
#include <hip/hip_runtime.h>
#include <cstdint>
#include <cstddef>

// ---------------------------------------------------------------------------
// ForwardDiffusionModel on MI455X (gfx1250, wave32)
//
// out[b,p] = a^250 * x0[b,p] + b_c * sum_t a^(249-t) * dist[b, rnd(b,p,t)]
// computed as the literal 250-step recurrence. Strategy:
//   * one workgroup per (batch, pixel-tile); the batch's FULL 65536-float
//     (256 KB) distribution table is staged into LDS -- only possible on
//     CDNA5's 320 KB-per-WGP LDS.
//   * LDS fill uses the CDNA5 async global->LDS data mover path
//     (global_load_async_to_lds_b128, ASYNCcnt, s_wait_asynccnt).
//   * 250-step loop runs out of registers with ds_load_b32 gathers; 4
//     independent chains per thread hide LDS latency.
// ---------------------------------------------------------------------------

#define STEPS            250
#define BATCHES          8
#define PIX_PER_BATCH    (1024 * 1024)
#define NDIST            65536                       // table entries per batch
#define THREADS          1024                        // 32 wave32s / WGP
#define EPT              4                           // elements per thread
#define ELEMS_PER_BLOCK  (THREADS * EPT)             // 4096
#define BLOCKS_PER_BATCH (PIX_PER_BATCH / ELEMS_PER_BLOCK)  // 256
#define LDS_BYTES        (NDIST * sizeof(float))     // 262144 B <= 320 KB WGP LDS

// sqrt(1-0.02), sqrt(0.02)
#define COEF_A 0.98994949366116653f
#define COEF_B 0.14142135623730950f

// Cheap counter-based hash (PCG output permutation). Deterministic per
// (element, step) -> same result on every graph replay.
__device__ __forceinline__ uint32_t pcg_hash(uint32_t v) {
    uint32_t state = v * 747796405u + 2891336453u;
    uint32_t word  = ((state >> ((state >> 28u) + 4u)) ^ state) * 277803737u;
    return (word >> 22u) ^ word;
}

__global__ __launch_bounds__(THREADS, 1)
void fwd_diffusion_kernel(const float* __restrict__ x0,     // [8, 1024*1024]
                          const float* __restrict__ dist,   // [8, 65536]
                          float* __restrict__ out) {        // [8, 1024*1024]
    extern __shared__ float sdist[];                 // 65536 floats = 256 KB

    const uint32_t batch = blockIdx.y;
    const float* dist_b  = dist + (size_t)batch * NDIST;

    // ------------------------------------------------------------------
    // Stage the whole 256 KB table into LDS with async global->LDS B128
    // copies (CDNA5 async-tensor path, tracked by ASYNCcnt).
    // 65536 floats / (1024 threads * 4 floats per b128) = 16 issues/thread.
    // Lane-relative LDS dest offset == global byte offset (same layout).
    // ------------------------------------------------------------------
    const uint32_t lds_base = (uint32_t)(uintptr_t)&sdist[0];
    #pragma unroll
    for (int i = 0; i < NDIST / (THREADS * 4); ++i) {
        uint32_t goff  = (threadIdx.x + (uint32_t)i * THREADS) * 16u; // byte off
        uint32_t laddr = lds_base + goff;
        asm volatile("global_load_async_to_lds_b128 %0, %1, %2"
                     :: "v"(laddr), "v"(goff), "s"(dist_b)
                     : "memory");
    }
    asm volatile("s_wait_asynccnt 0" ::: "memory");
    __syncthreads();

    // ------------------------------------------------------------------
    // 250-step recurrence, EPT independent chains per thread for ILP.
    // Coalesced global accesses: element p = blk*4096 + e*1024 + tid.
    // ------------------------------------------------------------------
    const size_t   batch_off = (size_t)batch * PIX_PER_BATCH;
    const uint32_t base_pix  = blockIdx.x * ELEMS_PER_BLOCK + threadIdx.x;

    float    x[EPT];
    uint32_t seed[EPT];
    #pragma unroll
    for (int e = 0; e < EPT; ++e) {
        uint32_t pix = base_pix + (uint32_t)e * THREADS;
        seed[e] = (batch * PIX_PER_BATCH + pix) * 0x9E3779B1u;   // element key
        x[e]    = x0[batch_off + pix];
    }

    for (int t = 0; t < STEPS; ++t) {
        const uint32_t tk = (uint32_t)t * 0x85EBCA6Bu;
        #pragma unroll
        for (int e = 0; e < EPT; ++e) {
            uint32_t h   = pcg_hash(seed[e] + tk);
            uint32_t idx = h & (NDIST - 1u);          // uniform in [0, 65536)
            float    s   = sdist[idx];                // ds_load_b32 gather
            x[e] = __builtin_fmaf(COEF_A, x[e], COEF_B * s);
        }
    }

    #pragma unroll
    for (int e = 0; e < EPT; ++e) {
        uint32_t pix = base_pix + (uint32_t)e * THREADS;
        out[batch_off + pix] = x[e];
    }
}

extern "C" void kernel_launch(void* const* d_in, const int* in_sizes, int n_in,
                              void* d_out, int out_size, void* d_ws, size_t ws_size,
                              hipStream_t stream) {
    const float* x0   = (const float*)d_in[0];   // reflectance_normal [8,1024,1024]
    const float* dist = (const float*)d_in[1];   // distribution       [8,65536]
    float*       out  = (float*)d_out;           // [8,1024,1024] f32

    // Allow >64 KB dynamic LDS (256 KB table; CDNA5 WGP has 320 KB).
    (void)hipFuncSetAttribute((const void*)fwd_diffusion_kernel,
                              hipFuncAttributeMaxDynamicSharedMemorySize,
                              (int)LDS_BYTES);

    dim3 grid(BLOCKS_PER_BATCH, BATCHES);        // (256, 8)
    dim3 block(THREADS);                         // 1024 = 32 wave32s
    fwd_diffusion_kernel<<<grid, block, LDS_BYTES, stream>>>(x0, dist, out);
}